// ChebConv_82016695485139
// MI455X (gfx1250) — compile-verified
//
#include <hip/hip_runtime.h>

// ---------------------------------------------------------------------------
// ChebConv for MI455X (gfx1250): fp32 WMMA (16x16x4), LDS-staged weights,
// block-per-edge SpMM with hardware f32 atomics. Bandwidth-bound problem
// (~50 GFLOP vs several GB of traffic @ 23.3 TB/s) -> keep full fp32.
// ---------------------------------------------------------------------------

#define V_N    20000
#define NNZ_N  640000
#define B_N    16
#define CIN_N  128
#define COUT_N 128
#define R_N    4
#define F_N    (B_N * CIN_N)          // 2048

typedef __attribute__((ext_vector_type(2))) float v2f;
typedef __attribute__((ext_vector_type(8))) float v8f;

// ---------------------------------------------------------------------------
// 1) Transpose x (F=2048, V) row-major  ->  x0 (V, F) row-major.
//    x is (B, Cin, V) flat == (f = b*128+c, v), exactly a (2048, V) matrix.
//    32x32 LDS tile, 256 threads, fully coalesced both directions.
//    V % 32 == 0 and F % 32 == 0, so no bounds checks (EXEC stays full).
// ---------------------------------------------------------------------------
__global__ __launch_bounds__(256) void k_transpose(const float* __restrict__ x,
                                                   float* __restrict__ y) {
    __shared__ float tile[32][33];
    const int vb = blockIdx.x * 32;     // v tile origin
    const int fb = blockIdx.y * 32;     // f tile origin
    const int tx = threadIdx.x & 31;
    const int ty = threadIdx.x >> 5;    // 0..7
#pragma unroll
    for (int i = 0; i < 32; i += 8) {
        tile[ty + i][tx] = x[(size_t)(fb + ty + i) * V_N + (vb + tx)];
    }
    __syncthreads();
#pragma unroll
    for (int i = 0; i < 32; i += 8) {
        y[(size_t)(vb + ty + i) * F_N + (fb + tx)] = tile[tx][ty + i];
    }
}

// ---------------------------------------------------------------------------
// 2) Seed kernel: dst = alpha * src (alpha==0 -> pure zero fill, no read).
//    Used to fold the Chebyshev recurrence (dst = -x_{k-2}) into the SpMM.
// ---------------------------------------------------------------------------
__global__ __launch_bounds__(256) void k_seed(const float4* __restrict__ src,
                                              float4* __restrict__ dst,
                                              float alpha, int n4) {
    const int i = blockIdx.x * 256 + threadIdx.x;
    if (i >= n4) return;
    if (alpha == 0.0f) {
        dst[i] = make_float4(0.f, 0.f, 0.f, 0.f);
    } else {
        float4 s = src[i];
        dst[i] = make_float4(alpha * s.x, alpha * s.y, alpha * s.z, alpha * s.w);
    }
}

// ---------------------------------------------------------------------------
// 3) SpMM: yout[rows[e], :] += scale * vals[e] * yin[cols[e], :]
//    One 256-thread block per edge; float4 gathers (rows ~L2-resident:
//    164 MB vs 192 MB L2, ~32x reuse per row) + scalar f32 atomics.
// ---------------------------------------------------------------------------
__global__ __launch_bounds__(256) void k_spmm(const float* __restrict__ yin,
                                              float* __restrict__ yout,
                                              const float* __restrict__ vals,
                                              const int* __restrict__ rows,
                                              const int* __restrict__ cols,
                                              float scale) {
    const int e = blockIdx.x;
    const float a = scale * vals[e];
    const float4* __restrict__ src =
        (const float4*)(yin + (size_t)cols[e] * F_N);
    float* __restrict__ dst = yout + (size_t)rows[e] * F_N;
    const int t = threadIdx.x;
#pragma unroll
    for (int half = 0; half < 2; ++half) {
        const int q = t + half * 256;        // float4 index, 0..511
        const float4 s = src[q];
        float* d = dst + q * 4;
        unsafeAtomicAdd(d + 0, a * s.x);
        unsafeAtomicAdd(d + 1, a * s.y);
        unsafeAtomicAdd(d + 2, a * s.z);
        unsafeAtomicAdd(d + 3, a * s.w);
    }
}

// ---------------------------------------------------------------------------
// 4) WMMA GEMM (one r at a time): out(b,co,v) += A(v,b,:) @ W(:,co)
//    A = basis matrix (V, 2048) row-major; per vertex v it is a 16(b) x 128(ci)
//    row-major tile with stride 128 -> exactly the M x K slab for WMMA.
//    One wave per vertex: 8 accumulators v8f cover the full 16x128 output.
//    W (64 KB) staged in LDS packed as (k-pair, co) float2 so each B fragment
//    of V_WMMA_F32_16X16X4_F32 is one ds_load_b64.
//
//    ISA layouts (CDNA5 ISA):
//      A 16x4 f32 : lane L<16 -> M=L, K=k+{0,1}; L>=16 -> M=L-16, K=k+{2,3}
//      B 4x16 f32 : lane L<16 -> N=L, K=k+{0,1}; L>=16 -> N=L-16, K=k+{2,3}
//      D 16x16    : vgpr j, lane L -> M = j + (L>=16)*8, N = L%16
// ---------------------------------------------------------------------------
__global__ __launch_bounds__(256) void k_gemm_wmma(const float* __restrict__ A,
                                                   const float* __restrict__ Wr,
                                                   const float* __restrict__ bias,
                                                   float* __restrict__ out,
                                                   int initFlag) {
    // sWp[k2*128 + co] = { W[2*k2][co], W[2*k2+1][co] }   (64 KB)
    __shared__ v2f sWp[(CIN_N / 2) * COUT_N];

    const int t    = threadIdx.x;
    const int wave = t >> 5;
    const int lane = t & 31;
    const int lh   = lane >> 4;     // lane half: 0 or 1
    const int ll   = lane & 15;

    // Cooperative pack of the 128x128 weight slice into LDS.
    for (int idx = t; idx < CIN_N * COUT_N; idx += 256) {
        const int k  = idx >> 7;
        const int co = idx & 127;
        sWp[(k >> 1) * COUT_N + co][k & 1] = Wr[idx];
    }
    __syncthreads();

    const int v = blockIdx.x * 8 + wave;              // vertex (grid exact)
    const float* __restrict__ Abase =
        A + (size_t)v * F_N + (size_t)ll * CIN_N + lh * 2;

    v8f acc[8];
    const v8f zero = {0.f, 0.f, 0.f, 0.f, 0.f, 0.f, 0.f, 0.f};
#pragma unroll
    for (int n = 0; n < 8; ++n) acc[n] = zero;

    // K = 128 in steps of 4; 8 independent accumulator chains hide the
    // WMMA->WMMA RAW latency (compiler inserts required NOPs otherwise).
#pragma unroll 4
    for (int k = 0; k < CIN_N; k += 4) {
        const v2f a = *(const v2f*)(Abase + k);           // contiguous 8B
        const int k2 = (k >> 1) + lh;                     // B row-pair idx
#pragma unroll
        for (int n = 0; n < 8; ++n) {
            const v2f b = sWp[k2 * COUT_N + n * 16 + ll]; // ds_load_b64
            acc[n] = __builtin_amdgcn_wmma_f32_16x16x4_f32(
                false, a, false, b, (short)0, acc[n], false, false);
        }
    }

    // Scatter D into out (B, Cout, V): element (vgpr j, lane) -> b = j+lh*8,
    // co = n*16+ll. r==0 seeds with bias, r>0 accumulates (tile ownership is
    // exclusive per wave, so plain RMW is race-free).
#pragma unroll
    for (int n = 0; n < 8; ++n) {
        const int co = n * 16 + ll;
        const float bval = bias[co];
#pragma unroll
        for (int j = 0; j < 8; ++j) {
            const int b = j + lh * 8;
            const size_t idx = ((size_t)b * COUT_N + co) * (size_t)V_N + v;
            const float base = initFlag ? bval : out[idx];
            out[idx] = base + acc[n][j];
        }
    }
}

// ---------------------------------------------------------------------------
// Host-side launch sequence. Workspace: 3 rotating basis buffers of V*F fp32
// (3 x 164 MB). Buffer rotation:
//   buf0 = x0; buf1 = L x0 (=x1); buf2 = 2 L x1 - x0 (=x2); buf0 = 2 L x2 - x1
// GEMM for each r accumulates straight into d_out.
// ---------------------------------------------------------------------------
extern "C" void kernel_launch(void* const* d_in, const int* in_sizes, int n_in,
                              void* d_out, int out_size, void* d_ws, size_t ws_size,
                              hipStream_t stream) {
    const float* x      = (const float*)d_in[0];   // (B, Cin, V)
    const float* weight = (const float*)d_in[1];   // (R, Cin, Cout)
    const float* bias   = (const float*)d_in[2];   // (Cout,)
    const float* vals   = (const float*)d_in[3];   // (NNZ,)
    const int*   rows   = (const int*)d_in[4];
    const int*   cols   = (const int*)d_in[5];
    float* out = (float*)d_out;                    // (B, Cout, V)

    const size_t S = (size_t)V_N * F_N;            // floats per basis buffer
    float* buf0 = (float*)d_ws;
    float* buf1 = buf0 + S;
    float* buf2 = buf1 + S;

    const int n4 = (int)(S / 4);
    const dim3 seedGrid((n4 + 255) / 256);
    const dim3 trGrid(V_N / 32, F_N / 32);
    const dim3 gemmGrid(V_N / 8);
    const dim3 spmmGrid(NNZ_N);
    const dim3 blk(256);

    // x0 = x^T  (V, 2048)
    k_transpose<<<trGrid, blk, 0, stream>>>(x, buf0);
    // out = bias + x0 @ W0
    k_gemm_wmma<<<gemmGrid, blk, 0, stream>>>(buf0, weight + 0 * CIN_N * COUT_N,
                                              bias, out, 1);
    // x1 = L x0
    k_seed<<<seedGrid, blk, 0, stream>>>((const float4*)buf1, (float4*)buf1, 0.0f, n4);
    k_spmm<<<spmmGrid, blk, 0, stream>>>(buf0, buf1, vals, rows, cols, 1.0f);
    k_gemm_wmma<<<gemmGrid, blk, 0, stream>>>(buf1, weight + 1 * CIN_N * COUT_N,
                                              bias, out, 0);
    // x2 = 2 L x1 - x0   (seed buf2 = -x0, SpMM adds with scale 2)
    k_seed<<<seedGrid, blk, 0, stream>>>((const float4*)buf0, (float4*)buf2, -1.0f, n4);
    k_spmm<<<spmmGrid, blk, 0, stream>>>(buf1, buf2, vals, rows, cols, 2.0f);
    k_gemm_wmma<<<gemmGrid, blk, 0, stream>>>(buf2, weight + 2 * CIN_N * COUT_N,
                                              bias, out, 0);
    // x3 = 2 L x2 - x1   (reuse buf0)
    k_seed<<<seedGrid, blk, 0, stream>>>((const float4*)buf1, (float4*)buf0, -1.0f, n4);
    k_spmm<<<spmmGrid, blk, 0, stream>>>(buf2, buf0, vals, rows, cols, 2.0f);
    k_gemm_wmma<<<gemmGrid, blk, 0, stream>>>(buf0, weight + 3 * CIN_N * COUT_N,
                                              bias, out, 0);
}